// CondConvMaskHead_83820581748830
// MI455X (gfx1250) — compile-verified
//
#include <hip/hip_runtime.h>
#include <math.h>

// ---------------------------------------------------------------------------
// CondInst dynamic mask head for MI455X (gfx1250, wave32).
// Phase 1: per-instance 3-layer MLP via V_WMMA_F32_16X16X4_F32 -> low-res
//          logits in d_ws (13.9 MB, stays L2-resident on 192 MB L2).
// Phase 2: aligned-bilinear x2 upsample + sigmoid -> d_out (55.7 MB, NT store).
// Bandwidth-bound (~70 MB HBM traffic); WMMA makes the MLP compute negligible
// while keeping full fp32 precision (matches reference dtype).
// ---------------------------------------------------------------------------

typedef __attribute__((ext_vector_type(2))) float v2f;
typedef __attribute__((ext_vector_type(8))) float v8f;

#define CH 8
#define CIN 8
#define NPARAMS 169
#define H_LO 136
#define W_LO 200
#define HW_LO (H_LO * W_LO)

__global__ __launch_bounds__(256) void condinst_mlp_wmma(
    const float* __restrict__ mask_feats,   // (N, 8, H, W)
    const float* __restrict__ params,       // (n_inst, 169)
    const float* __restrict__ locations,    // (n_inst, 2)
    const int*   __restrict__ im_inds,      // (n_inst,)
    const int*   __restrict__ fpn_levels,   // (n_inst,)
    const float* __restrict__ soi_tab,      // (5,)
    const int*   __restrict__ stride_p,     // scalar
    float*       __restrict__ logits_lo)    // (n_inst, HW_LO)
{
  const int inst = blockIdx.y;
  const int lane = threadIdx.x & 31;
  const int wave = threadIdx.x >> 5;
  const int tileBase = blockIdx.x * 256 + wave * 32;   // 32 pixels per wave

  const int   stride  = stride_p[0];
  const int   half    = stride >> 1;                    // stride // 2
  const float inv_soi = 1.0f / soi_tab[fpn_levels[inst]];
  const float locx = locations[2 * inst + 0];
  const float locy = locations[2 * inst + 1];
  const float* feats_im = mask_feats + (size_t)im_inds[inst] * CIN * HW_LO;
  const float* pp       = params + (size_t)inst * NPARAMS;

  const int  colInLane = lane & 15;
  const int  rowSel    = (lane < 16) ? 0 : 2;   // A/B lane-half K offset
  const int  rA        = lane & 7;              // weight row, M=8..15 duplicate 0..7
  const bool lo        = (lane < 16);

  // ---- A fragments (weights; rows 8..15 duplicated so the C-layout output
  //      is directly consumable as the next layer's B fragments) ----
  v2f a0[3];
#pragma unroll
  for (int c = 0; c < 3; ++c) {
#pragma unroll
    for (int j = 0; j < 2; ++j) {
      int k = 4 * c + j + rowSel;                       // K index, pad 10->12
      float w = pp[rA * 10 + ((k < 10) ? k : 9)];       // w0: (8,10) row-major
      a0[c][j] = (k < 10) ? w : 0.0f;
    }
  }
  v2f a1[2];
#pragma unroll
  for (int c = 0; c < 2; ++c) {
#pragma unroll
    for (int j = 0; j < 2; ++j) {
      int k = 4 * c + j + rowSel;                       // always < 8
      a1[c][j] = pp[80 + rA * 8 + k];                   // w1: (8,8)
    }
  }

  // uniform (scalarized) bias / final-layer weights
  float b0v[8], b1v[8], w2v[8];
#pragma unroll
  for (int v = 0; v < 8; ++v) {
    b0v[v] = pp[152 + v];
    b1v[v] = pp[160 + v];
    w2v[v] = pp[144 + v];
  }
  const float b2 = pp[168];

  v8f acc[2];  // post-ReLU layer-1 outputs for the two 16-pixel N tiles
#pragma unroll
  for (int t = 0; t < 2; ++t) {
    const int p  = tileBase + 16 * t + colInLane;
    const int pc = (p < HW_LO) ? p : (HW_LO - 1);       // clamp loads only

    // rel coords computed once per tile (branchless select later)
    const int   px   = pc % W_LO;
    const int   py   = pc / W_LO;
    const float relx = (locx - (float)(px * stride + half)) * inv_soi;
    const float rely = (locy - (float)(py * stride + half)) * inv_soi;

    // ---- layer-0 B fragments, fully branchless: always issue the feature
    //      load with a clamped channel (L1/L2 hit), then v_cndmask-select.
    //      Keeps EXEC untouched and lets the loads form one clause. ----
    v2f b[3];
#pragma unroll
    for (int c = 0; c < 3; ++c) {
#pragma unroll
      for (int j = 0; j < 2; ++j) {
        int row = 4 * c + j + rowSel;                   // 0..11, per lane-half
        int ch  = row - 2;
        ch = (ch < 0) ? 0 : ((ch > 7) ? 7 : ch);
        float f   = feats_im[ch * HW_LO + pc];          // unconditional load
        float rel = (row == 0) ? relx : rely;
        float v   = (row < 2) ? rel : ((row < 10) ? f : 0.0f);
        b[c][j] = v;
      }
    }

    // ---- layer 0: C init = bias (rows duplicated -> uniform per VGPR) ----
    v8f c0;
#pragma unroll
    for (int v = 0; v < 8; ++v) c0[v] = b0v[v];
#pragma unroll
    for (int c = 0; c < 3; ++c)
      c0 = __builtin_amdgcn_wmma_f32_16x16x4_f32(false, a0[c], false, b[c],
                                                 (short)0, c0, false, false);
#pragma unroll
    for (int v = 0; v < 8; ++v) c0[v] = fmaxf(c0[v], 0.0f);  // ReLU

    // ---- layer 1: B fragments fall straight out of the duplicated-row
    //      C layout (v_cndmask only, no cross-lane traffic) ----
    v2f bb0, bb1;
    bb0[0] = lo ? c0[0] : c0[2];   // K=0 | K=2
    bb0[1] = lo ? c0[1] : c0[3];   // K=1 | K=3
    bb1[0] = lo ? c0[4] : c0[6];   // K=4 | K=6
    bb1[1] = lo ? c0[5] : c0[7];   // K=5 | K=7

    v8f c1;
#pragma unroll
    for (int v = 0; v < 8; ++v) c1[v] = b1v[v];
    c1 = __builtin_amdgcn_wmma_f32_16x16x4_f32(false, a1[0], false, bb0,
                                               (short)0, c1, false, false);
    c1 = __builtin_amdgcn_wmma_f32_16x16x4_f32(false, a1[1], false, bb1,
                                               (short)0, c1, false, false);
#pragma unroll
    for (int v = 0; v < 8; ++v) c1[v] = fmaxf(c1[v], 0.0f);  // ReLU
    acc[t] = c1;
  }

  // ---- layer 2 (1x8 dot) in VALU: every lane owns all 8 channels of its
  //      pixel column thanks to the duplication -> 32 coalesced stores ----
  float logit = b2;
#pragma unroll
  for (int r = 0; r < 8; ++r) {
    float s = lo ? acc[0][r] : acc[1][r];
    logit = fmaf(w2v[r], s, logit);
  }
  const int pixel = tileBase + lane;
  if (pixel < HW_LO)
    logits_lo[(size_t)inst * HW_LO + pixel] = logit;
}

// ---------------------------------------------------------------------------
// Phase 2: aligned_bilinear(factor=2) + sigmoid.
// Static weights: o==0 -> L[0]; o odd -> L[(o-1)/2]; o even -> avg of o/2-1,o/2.
// ---------------------------------------------------------------------------
__global__ __launch_bounds__(256) void condinst_upsample_sigmoid(
    const float* __restrict__ logits_lo, float* __restrict__ out, int total)
{
  int idx = blockIdx.x * 256 + threadIdx.x;
  if (idx >= total) return;
  const int OW = 2 * W_LO;
  const int OHW = 4 * HW_LO;
  int i  = idx / OHW;
  int r  = idx % OHW;
  int oy = r / OW, ox = r % OW;

  int ly0, ly1; float fy;
  if (oy == 0)      { ly0 = 0;             ly1 = 0;       fy = 0.0f; }
  else if (oy & 1)  { ly0 = (oy - 1) >> 1; ly1 = ly0;     fy = 0.0f; }
  else              { ly1 = oy >> 1;       ly0 = ly1 - 1; fy = 0.5f; }
  int lx0, lx1; float fx;
  if (ox == 0)      { lx0 = 0;             lx1 = 0;       fx = 0.0f; }
  else if (ox & 1)  { lx0 = (ox - 1) >> 1; lx1 = lx0;     fx = 0.0f; }
  else              { lx1 = ox >> 1;       lx0 = lx1 - 1; fx = 0.5f; }

  const float* base = logits_lo + (size_t)i * HW_LO;
  float v00 = base[ly0 * W_LO + lx0];
  float v01 = base[ly0 * W_LO + lx1];
  float v10 = base[ly1 * W_LO + lx0];
  float v11 = base[ly1 * W_LO + lx1];
  float top = v00 + fx * (v01 - v00);
  float bot = v10 + fx * (v11 - v10);
  float val = top + fy * (bot - top);
  float sig = 1.0f / (1.0f + __expf(-val));
  __builtin_nontemporal_store(sig, out + idx);   // streamed output, keep out of L2
}

// ---------------------------------------------------------------------------
extern "C" void kernel_launch(void* const* d_in, const int* in_sizes, int n_in,
                              void* d_out, int out_size, void* d_ws, size_t ws_size,
                              hipStream_t stream) {
  const float* mask_feats = (const float*)d_in[0];
  const float* params     = (const float*)d_in[1];
  const float* locations  = (const float*)d_in[2];
  const int*   im_inds    = (const int*)d_in[3];
  const int*   fpn_levels = (const int*)d_in[4];
  const float* soi_tab    = (const float*)d_in[5];
  const int*   stride_p   = (const int*)d_in[6];
  float*       out        = (float*)d_out;
  float*       lo         = (float*)d_ws;       // n_inst*HW_LO floats (13.9 MB)

  const int n_inst = in_sizes[1] / NPARAMS;     // 128

  dim3 g1((HW_LO + 255) / 256, n_inst);
  condinst_mlp_wmma<<<g1, 256, 0, stream>>>(mask_feats, params, locations,
                                            im_inds, fpn_levels, soi_tab,
                                            stride_p, lo);

  const int total = out_size;                   // n_inst * (2H) * (2W)
  condinst_upsample_sigmoid<<<(total + 255) / 256, 256, 0, stream>>>(lo, out,
                                                                     total);
}